// FeatureAggregator_74062416053446
// MI455X (gfx1250) — compile-verified
//
#include <hip/hip_runtime.h>
#include <hip/hip_bf16.h>
#include <stdint.h>

// Problem dims (fixed by reference setup_inputs):
//   embeddings [B=4096, N=128, D=512] f32, mask [B, N] i32, out [B, D] f32
#define N_ROWS   128
#define D_DIM    512
#define THREADS  128                 // one float4 of D per thread
#define TILE_N   8                   // rows per LDS tile
#define NTILES   (N_ROWS / TILE_N)   // 16
#define ROW_BYTES (D_DIM * 4)        // 2048 B per row
#define TILE_BYTES (TILE_N * ROW_BYTES)   // 16 KB per buffer
#define TILE_F4  (TILE_BYTES / 16)        // 1024 float4

// CDNA5 async copy tile: global -> LDS, 16B/lane/op, tracked by ASYNCcnt.
// IOFFSET is added to BOTH the LDS (VDST) and global (VADDR) addresses
// (ISA 08 async pseudocode), and our row stride is identical on both sides,
// so one base address pair + literal offsets covers the whole 8-row tile.
#define ASYNC_ROW(ldsaddr, gptr, OFS_TXT)                                       \
    asm volatile("global_load_async_to_lds_b128 %0, %1, off" OFS_TXT            \
                 " th:TH_LOAD_NT"                                               \
                 :: "v"(ldsaddr), "v"(gptr) : "memory")

#define ASYNC_TILE_LD(ldsaddr, gptr)                 \
    do {                                             \
        ASYNC_ROW(ldsaddr, gptr, "");                \
        ASYNC_ROW(ldsaddr, gptr, " offset:2048");    \
        ASYNC_ROW(ldsaddr, gptr, " offset:4096");    \
        ASYNC_ROW(ldsaddr, gptr, " offset:6144");    \
        ASYNC_ROW(ldsaddr, gptr, " offset:8192");    \
        ASYNC_ROW(ldsaddr, gptr, " offset:10240");   \
        ASYNC_ROW(ldsaddr, gptr, " offset:12288");   \
        ASYNC_ROW(ldsaddr, gptr, " offset:14336");   \
    } while (0)

__global__ __launch_bounds__(THREADS)
void FeatureAggregator_74062416053446_kernel(const float* __restrict__ emb,
                                             const int* __restrict__ mask,
                                             float* __restrict__ out) {
    __shared__ float4 tiles[2][TILE_F4];   // double buffer, 32 KB
    __shared__ int    mrow[N_ROWS];        // mask row cache, 512 B

    const int b   = blockIdx.x;
    const int tid = threadIdx.x;

    // Per-thread base addresses. Generic LDS pointer truncated to 32 bits
    // yields the LDS byte offset (ISA 10.2 aperture mapping).
    const char* gbase = (const char*)(emb + (size_t)b * N_ROWS * D_DIM) + tid * 16;
    const uint32_t lds0 = (uint32_t)(uintptr_t)(&tiles[0][0]) + tid * 16;
    const uint32_t lds1 = (uint32_t)(uintptr_t)(&tiles[1][0]) + tid * 16;

    // Kick off tile 0 DMA first so the HBM stream starts immediately
    // (8 async B128 per wave -> ASYNCcnt = 8).
    ASYNC_TILE_LD(lds0, gbase);

    // Stage this batch's mask row in LDS (one int per thread).
    mrow[tid] = mask[b * N_ROWS + tid];
    __syncthreads();  // make mrow[] visible to all threads

    float4 mx = make_float4(-3.0e38f, -3.0e38f, -3.0e38f, -3.0e38f);
    float4 mn = make_float4( 3.0e38f,  3.0e38f,  3.0e38f,  3.0e38f);

    const float4* lbuf0 = &tiles[0][0];
    const float4* lbuf1 = &tiles[1][0];
    const char*   gnext = gbase + TILE_BYTES;   // tile t+1 base, bumped per iter

    for (int t = 0; t < NTILES; ++t) {
        if (t + 1 < NTILES) {
            // WAR guard: the buffer we are about to DMA-overwrite was ds-read
            // last iteration; make sure those LDS reads have retired.
            asm volatile("s_wait_dscnt 0" ::: "memory");
            ASYNC_TILE_LD(((t + 1) & 1) ? lds1 : lds0, gnext);
            gnext += TILE_BYTES;
            // 8 ops outstanding from tile t+1 => asynccnt <= 8 means tile t done.
            asm volatile("s_wait_asynccnt 8" ::: "memory");
        } else {
            asm volatile("s_wait_asynccnt 0" ::: "memory");
        }

        const float4* buf = (t & 1) ? lbuf1 : lbuf0;
#pragma unroll
        for (int j = 0; j < TILE_N; ++j) {
            // Each thread consumes exactly the bytes its own wave DMA'd: no barrier.
            float4 v = buf[j * THREADS + tid];
            // Mask bit is uniform across the block -> hoist to SGPR, scalar branch
            // skips all 8 VALU updates for invalid rows.
            int m = __builtin_amdgcn_readfirstlane(mrow[t * TILE_N + j]);
            if (m) {
                mx.x = fmaxf(mx.x, v.x); mx.y = fmaxf(mx.y, v.y);
                mx.z = fmaxf(mx.z, v.z); mx.w = fmaxf(mx.w, v.w);
                mn.x = fminf(mn.x, v.x); mn.y = fminf(mn.y, v.y);
                mn.z = fminf(mn.z, v.z); mn.w = fminf(mn.w, v.w);
            }
        }
    }

    float4 r;
    r.x = mx.x - mn.x;
    r.y = mx.y - mn.y;
    r.z = mx.z - mn.z;
    r.w = mx.w - mn.w;
    ((float4*)out)[(size_t)b * THREADS + tid] = r;  // coalesced B128 store
}

extern "C" void kernel_launch(void* const* d_in, const int* in_sizes, int n_in,
                              void* d_out, int out_size, void* d_ws, size_t ws_size,
                              hipStream_t stream) {
    const float* emb  = (const float*)d_in[0];  // [B, N, D] f32
    const int*   mask = (const int*)d_in[1];    // [B, N] i32
    float*       out  = (float*)d_out;          // [B, D] f32

    // B derived from mask size so the launch adapts if B changes.
    const int B = in_sizes[1] / N_ROWS;

    FeatureAggregator_74062416053446_kernel<<<B, THREADS, 0, stream>>>(emb, mask, out);
}